// FlowProjectionModule_35545149341801
// MI455X (gfx1250) — compile-verified
//
#include <hip/hip_runtime.h>

// DAIN-style flow projection (forward splatting) for flow [B, 2, 1080, 1920] f32.
// out[:,0]=avg(-u), out[:,1]=avg(-v) splatted to the 4 integer neighbors of
// (x+u, y+v), averaged by hit count, 0 where count==0.
//
// Scatter-with-atomics: no matrix structure -> WMMA inapplicable. Binder is
// L2 atomic RMW throughput (199M f32 atomics naive). Strategy: aggregate each
// 32x32 source tile's splats in LDS (ds_add_f32, +/-6 halo window), flush the
// window once with global atomics (5.7/px instead of 12/px), with a direct
// global-atomic fallback for the rare out-of-window displacement.

#define HH 1080
#define WW 1920

#define TW   32                 // tile width  (source pixels)
#define THt  32                 // tile height (source pixels)
#define PAD  6                  // halo in each direction
#define WINW (TW + 2 * PAD)     // 44
#define WINH (THt + 2 * PAD)    // 44
#define NCELL (WINW * WINH)     // 1936

// ---------------------------------------------------------------- zero pass
__global__ void fp_zero_kernel(float* __restrict__ out, float* __restrict__ cnt,
                               int n_out4, int n_cnt4) {
  int i = blockIdx.x * blockDim.x + threadIdx.x;
  float4 z = make_float4(0.f, 0.f, 0.f, 0.f);
  if (i < n_out4) reinterpret_cast<float4*>(out)[i] = z;
  if (i < n_cnt4) reinterpret_cast<float4*>(cnt)[i] = z;
}

// ---------------------------------------------------------------- splat pass
// Block = 256 threads; each thread owns 4 consecutive pixels of one row of a
// 32x32 source tile. grid = (W/TW, ceil(H/THt), B).
__global__ __launch_bounds__(256) void fp_splat_tiled(
    const float* __restrict__ flow,
    float* __restrict__ acc,    // == d_out, [B,2,H,W]
    float* __restrict__ cnt) {  // [B,H,W]
  __shared__ float s_u[NCELL];
  __shared__ float s_v[NCELL];
  __shared__ float s_c[NCELL];

  const int HW = HH * WW;
  const int b  = blockIdx.z;
  const int x0 = blockIdx.x * TW;
  const int y0 = blockIdx.y * THt;
  const int t  = threadIdx.x;

  // zero the LDS accumulation window
  for (int i = t; i < NCELL; i += 256) {
    s_u[i] = 0.f; s_v[i] = 0.f; s_c[i] = 0.f;
  }
  __syncthreads();

  const float* up = flow + (size_t)b * 2 * HW;   // u plane
  const float* vp = up + HW;                     // v plane
  float* accu = acc + (size_t)b * 2 * HW;        // out[:,0] plane
  float* accv = accu + HW;                       // out[:,1] plane
  float* cntp = cnt + (size_t)b * HW;

  const int lx = (t & 7) * 4;   // 0,4,...,28
  const int ly = t >> 3;        // 0..31
  const int x  = x0 + lx;
  const int y  = y0 + ly;

  if (y < HH) {
    int r = y * WW + x;
    float4 u4 = *reinterpret_cast<const float4*>(up + r);
    float4 v4 = *reinterpret_cast<const float4*>(vp + r);
    float us[4] = {u4.x, u4.y, u4.z, u4.w};
    float vs[4] = {v4.x, v4.y, v4.z, v4.w};

#pragma unroll
    for (int k = 0; k < 4; ++k) {
      float u = us[k];
      float v = vs[k];
      float fx = (float)(x + k) + u;
      float fy = (float)y + v;
      int xf = (int)floorf(fx);
      int yf = (int)floorf(fy);
      float nu = -u;
      float nv = -v;
#pragma unroll
      for (int dy = 0; dy <= 1; ++dy) {
        int ty = yf + dy;
        int wy = ty - y0 + PAD;
#pragma unroll
        for (int dx = 0; dx <= 1; ++dx) {
          int tx = xf + dx;
          int wx = tx - x0 + PAD;
          if ((unsigned)wx < (unsigned)WINW && (unsigned)wy < (unsigned)WINH) {
            // common case: aggregate in LDS (image-bounds applied at flush)
            int o = wy * WINW + wx;
            atomicAdd(&s_u[o], nu);
            atomicAdd(&s_v[o], nv);
            atomicAdd(&s_c[o], 1.0f);
          } else if ((unsigned)tx < (unsigned)WW && (unsigned)ty < (unsigned)HH) {
            // rare large displacement: direct global atomics
            int o = ty * WW + tx;
            atomicAdd(accu + o, nu);
            atomicAdd(accv + o, nv);
            atomicAdd(cntp + o, 1.0f);
          }
        }
      }
    }
  }
  __syncthreads();

  // flush LDS window to global accumulators (tiles' halos overlap -> atomics)
  for (int i = t; i < NCELL; i += 256) {
    float c = s_c[i];
    if (c != 0.f) {
      int wy = i / WINW;
      int wx = i - wy * WINW;
      int gx = x0 - PAD + wx;
      int gy = y0 - PAD + wy;
      if ((unsigned)gx < (unsigned)WW && (unsigned)gy < (unsigned)HH) {
        int o = gy * WW + gx;
        atomicAdd(accu + o, s_u[i]);
        atomicAdd(accv + o, s_v[i]);
        atomicAdd(cntp + o, c);
      }
    }
  }
}

// ------------------------------------------------------------ normalize pass
__global__ void fp_norm_kernel(float* __restrict__ out,
                               const float* __restrict__ cnt, int nthreads) {
  const int HW = HH * WW;
  int t = blockIdx.x * blockDim.x + threadIdx.x;
  if (t >= nthreads) return;
  int g = t * 4;
  int b = g / HW;
  int r = g - b * HW;

  float* accu = out + (size_t)b * 2 * HW + r;
  float* accv = accu + HW;
  const float* cp = cnt + (size_t)b * HW + r;

  float4 c = *reinterpret_cast<const float4*>(cp);
  float4 u = *reinterpret_cast<const float4*>(accu);
  float4 v = *reinterpret_cast<const float4*>(accv);

  u.x = (c.x > 0.f) ? u.x / c.x : 0.f;
  u.y = (c.y > 0.f) ? u.y / c.y : 0.f;
  u.z = (c.z > 0.f) ? u.z / c.z : 0.f;
  u.w = (c.w > 0.f) ? u.w / c.w : 0.f;
  v.x = (c.x > 0.f) ? v.x / c.x : 0.f;
  v.y = (c.y > 0.f) ? v.y / c.y : 0.f;
  v.z = (c.z > 0.f) ? v.z / c.z : 0.f;
  v.w = (c.w > 0.f) ? v.w / c.w : 0.f;

  *reinterpret_cast<float4*>(accu) = u;
  *reinterpret_cast<float4*>(accv) = v;
}

extern "C" void kernel_launch(void* const* d_in, const int* in_sizes, int n_in,
                              void* d_out, int out_size, void* d_ws, size_t ws_size,
                              hipStream_t stream) {
  (void)n_in; (void)out_size; (void)ws_size;
  const float* flow = (const float*)d_in[0];
  float* out = (float*)d_out;
  float* cnt = (float*)d_ws;            // needs B*H*W*4 bytes (~66 MB)

  const int HW = HH * WW;
  int total = in_sizes[0];              // B*2*H*W
  int B = total / (2 * HW);
  int N = B * HW;                       // pixels

  dim3 blk(256);

  int n_out4 = (2 * N) / 4;
  int n_cnt4 = N / 4;
  fp_zero_kernel<<<(n_out4 + 255) / 256, blk, 0, stream>>>(out, cnt, n_out4, n_cnt4);

  dim3 sgrid(WW / TW, (HH + THt - 1) / THt, B);   // (60, 34, B)
  fp_splat_tiled<<<sgrid, blk, 0, stream>>>(flow, out, cnt);

  int nthreads = N / 4;
  fp_norm_kernel<<<(nthreads + 255) / 256, blk, 0, stream>>>(out, cnt, nthreads);
}